// MultiHeadManhattanAttention_84241488543814
// MI455X (gfx1250) — compile-verified
//
#include <hip/hip_runtime.h>
#include <math.h>

// Problem constants (match reference)
#define HIDDEN   512
#define HEADS    8
#define HEAD_DIM 64
#define BATCH    4
#define SEQ      2048
#define NTOK     (BATCH * SEQ)
#define EPS_LN   1e-5f

// Attention tiling
#define QT        16      // query rows per block
#define KT        64      // key rows staged in LDS per step
#define QK_STRIDE 68      // padded row stride (floats) for Q/K tiles (bank-conflict free)
#define SC_STRIDE 2052    // padded row stride (floats) for score rows (2048 + 4)

typedef __attribute__((ext_vector_type(2))) float v2f;
typedef __attribute__((ext_vector_type(8))) float v8f;
typedef int v4i_ __attribute__((vector_size(16)));   // matches async-LDS builtin param type

// ---- gfx1250 async global->LDS DMA (GLOBAL_LOAD_ASYNC_TO_LDS_B128) ----------
#if defined(__has_builtin)
#  if __has_builtin(__builtin_amdgcn_global_load_async_to_lds_b128)
#    define USE_ASYNC_LDS 1
#  endif
#endif
#ifndef USE_ASYNC_LDS
#  define USE_ASYNC_LDS 0
#endif

__device__ __forceinline__ void copy16_to_lds(const float* gsrc, float* ldst) {
#if USE_ASYNC_LDS
  __builtin_amdgcn_global_load_async_to_lds_b128((v4i_*)gsrc, (v4i_*)ldst, 0, 0);
#else
  *(float4*)ldst = *(const float4*)gsrc;
#endif
}
__device__ __forceinline__ void async_lds_wait() {
#if USE_ASYNC_LDS
  asm volatile("s_wait_asynccnt 0x0" ::: "memory");
#endif
}

__device__ __forceinline__ float wave_max(float v) {
#pragma unroll
  for (int off = 16; off > 0; off >>= 1) v = fmaxf(v, __shfl_xor(v, off, 32));
  return v;
}
__device__ __forceinline__ float wave_sum(float v) {
#pragma unroll
  for (int off = 16; off > 0; off >>= 1) v += __shfl_xor(v, off, 32);
  return v;
}

// ---------------------------------------------------------------------------
// C[t,o] = sum_k X[t,k] * W[o,k] + bias[o]   (i.e. X @ W^T + b)
// One wave computes one 16x16 output tile via V_WMMA_F32_16X16X4_F32.
// mode 0: scatter output into per-head layout [b][h][s][d]  (QKV projections)
// mode 1: flat [t][o] plus residual add                     (output projection)
// ---------------------------------------------------------------------------
__global__ __launch_bounds__(128)
void proj_gemm_kernel(const float* __restrict__ X, const float* __restrict__ W,
                      const float* __restrict__ bias, const float* __restrict__ resid,
                      float* __restrict__ out, int mode)
{
  const int lane  = threadIdx.x & 31;
  const int wid   = threadIdx.x >> 5;
  const int tile  = blockIdx.x * 4 + wid;   // 512*32 = 16384 tiles total
  const int otile = tile & 31;              // 32 column tiles (HIDDEN/16)
  const int ttile = tile >> 5;              // 512 row tiles (NTOK/16)
  const int row0  = ttile * 16;
  const int col0  = otile * 16;
  const int hlf   = lane >> 4;              // 0 or 1
  const int ll    = lane & 15;

  // A-fragment (16x4 f32): lanes 0-15 hold row ll K={0,1}; lanes 16-31 K={2,3}
  // B-fragment (4x16 f32): lane ll holds column ll; rows K split by lane half
  const float* xrow = X + (size_t)(row0 + ll) * HIDDEN;
  const float* wrow = W + (size_t)(col0 + ll) * HIDDEN;

  v8f acc = {};
  for (int k = 0; k < HIDDEN; k += 4) {
    v2f a = *(const v2f*)(xrow + k + 2 * hlf);
    v2f b = *(const v2f*)(wrow + k + 2 * hlf);
    acc = __builtin_amdgcn_wmma_f32_16x16x4_f32(false, a, false, b,
                                                (short)0, acc, false, false);
  }

  const int   col  = col0 + ll;
  const float bcol = bias[col];
#pragma unroll
  for (int r = 0; r < 8; ++r) {
    const int   row = row0 + r + 8 * hlf;   // token index (C layout: VGPR r, lane half)
    const float v   = acc[r] + bcol;
    if (mode == 0) {
      const int bb = row >> 11;             // / SEQ
      const int ss = row & (SEQ - 1);
      const int hh = col >> 6;              // / HEAD_DIM
      const int dd = col & (HEAD_DIM - 1);
      out[(((size_t)(bb * HEADS + hh) * SEQ) + ss) * HEAD_DIM + dd] = v;
    } else {
      const size_t idx = (size_t)row * HIDDEN + col;
      out[idx] = v + resid[idx];
    }
  }
}

// ---------------------------------------------------------------------------
// L1 attention: one block = one (b,h) pair x 16 query rows.
// Phase 1: VALU L1 distances with Q/K tiles in LDS (scores kept in LDS).
//          Tiles staged via GLOBAL_LOAD_ASYNC_TO_LDS_B128 (ASYNCcnt tracked).
// Phase 2: in-LDS softmax (wave32 shuffle reductions), attn streamed to HBM
//          with nontemporal stores (write-only; keep L2 for V/K reuse).
// Phase 3: attn @ V via WMMA f32 16x16x4, A-fragments read straight from LDS.
// ---------------------------------------------------------------------------
__global__ __launch_bounds__(256)
void l1_attention_kernel(const float* __restrict__ Qh, const float* __restrict__ Kh,
                         const float* __restrict__ Vh, const int* __restrict__ mask,
                         const float* __restrict__ temperature,
                         float* __restrict__ attn_out, float* __restrict__ outH)
{
  extern __shared__ float smem[];
  float* q_s = smem;                        // QT * QK_STRIDE
  float* k_s = q_s + QT * QK_STRIDE;        // KT * QK_STRIDE (reused as reduce buf)
  float* sc  = k_s + KT * QK_STRIDE;        // QT * SC_STRIDE

  const int tid  = threadIdx.x;
  const int lane = tid & 31;
  const int wid  = tid >> 5;
  const int qt   = blockIdx.x & (SEQ / QT - 1);   // 128 query tiles
  const int bh   = blockIdx.x >> 7;
  const int b    = bh >> 3;
  const int h    = bh & 7;
  const float temp = temperature[h];
  const int i0   = qt * QT;

  const float* Qbase = Qh + (size_t)bh * SEQ * HEAD_DIM;
  const float* Kbase = Kh + (size_t)bh * SEQ * HEAD_DIM;
  const float* Vbase = Vh + (size_t)bh * SEQ * HEAD_DIM;
  const int*   mrow  = mask + b * SEQ;

  // --- async-load Q tile into LDS: 16 rows x 64 f32 = 256 x 16B chunks ---
  {
    const int i = tid >> 4;        // row 0..15
    const int x = tid & 15;        // 16B chunk within row
    copy16_to_lds(Qbase + (size_t)(i0 + i) * HEAD_DIM + x * 4,
                  q_s + i * QK_STRIDE + x * 4);
  }

  const int qi = tid & 15;   // this thread's query row
  const int jg = tid >> 4;   // 0..15 : key-group within tile

  // --- Phase 1: L1 distances -> masked scores in LDS ---
  for (int jt = 0; jt < SEQ; jt += KT) {
    // async-load K tile: 64 rows x 64 f32 = 1024 x 16B chunks, 4 per thread
    for (int c = tid; c < KT * (HEAD_DIM / 4); c += 256) {
      const int j = c >> 4;        // key row 0..63
      const int x = c & 15;        // 16B chunk within row
      copy16_to_lds(Kbase + (size_t)(jt + j) * HEAD_DIM + x * 4,
                    k_s + j * QK_STRIDE + x * 4);
    }
    async_lds_wait();
    __syncthreads();

    float acc0 = 0.f, acc1 = 0.f, acc2 = 0.f, acc3 = 0.f;
    const float* qrow = q_s + qi * QK_STRIDE;
    const float* kr0  = k_s + (jg +  0) * QK_STRIDE;
    const float* kr1  = k_s + (jg + 16) * QK_STRIDE;
    const float* kr2  = k_s + (jg + 32) * QK_STRIDE;
    const float* kr3  = k_s + (jg + 48) * QK_STRIDE;
#pragma unroll 4
    for (int d = 0; d < HEAD_DIM; d += 4) {
      const float4 q4 = *(const float4*)(qrow + d);
      const float4 kA = *(const float4*)(kr0 + d);
      const float4 kB = *(const float4*)(kr1 + d);
      const float4 kC = *(const float4*)(kr2 + d);
      const float4 kD = *(const float4*)(kr3 + d);
      acc0 += fabsf(q4.x-kA.x)+fabsf(q4.y-kA.y)+fabsf(q4.z-kA.z)+fabsf(q4.w-kA.w);
      acc1 += fabsf(q4.x-kB.x)+fabsf(q4.y-kB.y)+fabsf(q4.z-kB.z)+fabsf(q4.w-kB.w);
      acc2 += fabsf(q4.x-kC.x)+fabsf(q4.y-kC.y)+fabsf(q4.z-kC.z)+fabsf(q4.w-kC.w);
      acc3 += fabsf(q4.x-kD.x)+fabsf(q4.y-kD.y)+fabsf(q4.z-kD.z)+fabsf(q4.w-kD.w);
    }
    const int jb = jt + jg;
    float* srow = sc + qi * SC_STRIDE;
    srow[jb +  0] = (mrow[jb +  0] == 0) ? -1e9f : -acc0 * temp;
    srow[jb + 16] = (mrow[jb + 16] == 0) ? -1e9f : -acc1 * temp;
    srow[jb + 32] = (mrow[jb + 32] == 0) ? -1e9f : -acc2 * temp;
    srow[jb + 48] = (mrow[jb + 48] == 0) ? -1e9f : -acc3 * temp;
    __syncthreads();
  }

  // --- Phase 2: softmax per row (2 rows per wave), stream attn to HBM ---
  for (int r = wid; r < QT; r += 8) {
    float* srow = sc + r * SC_STRIDE;
    float m = -1e30f;
    for (int j = lane; j < SEQ; j += 32) m = fmaxf(m, srow[j]);
    m = wave_max(m);
    float s = 0.f;
    for (int j = lane; j < SEQ; j += 32) {
      const float e = __expf(srow[j] - m);
      srow[j] = e;
      s += e;
    }
    s = wave_sum(s);
    const float inv = 1.f / s;
    float* arow = attn_out + ((size_t)bh * SEQ + (i0 + r)) * SEQ;
    for (int j = lane; j < SEQ; j += 32) {
      const float a = srow[j] * inv;
      srow[j] = a;                          // keep normalized attn in LDS
      __builtin_nontemporal_store(a, arow + j);   // write-only stream to HBM
    }
  }
  __syncthreads();

  // --- Phase 3: out[16][64] = attn[16][2048] @ V[2048][64] via WMMA ---
  // 8 waves: ntile = wid&3 selects a 16-wide d slice; wid>>2 splits K in half.
  const int ntile = wid & 3;
  const int ks    = (wid >> 2) * (SEQ / 2);
  const int d0    = ntile * 16;
  const int hlf   = lane >> 4;
  const int ll    = lane & 15;

  v8f acc = {};
  const float* arow_l = sc + ll * SC_STRIDE + ks;          // A row ll from LDS
  const float* vcol   = Vbase + (size_t)ks * HEAD_DIM + d0 + ll;
  for (int k = 0; k < SEQ / 2; k += 4) {
    const v2f a = *(const v2f*)(arow_l + k + 2 * hlf);      // ds_load_b64
    const float* vp = vcol + (size_t)(k + 2 * hlf) * HEAD_DIM;
    if ((k & 31) == 0)
      __builtin_prefetch(vp + 32 * HEAD_DIM, 0, 1);         // global_prefetch_b8
    v2f bf;
    bf.x = vp[0];
    bf.y = vp[HEAD_DIM];
    acc = __builtin_amdgcn_wmma_f32_16x16x4_f32(false, a, false, bf,
                                                (short)0, acc, false, false);
  }

  // combine the two K-halves through LDS (reuse k_s region: needs 4*256 floats)
  float* red = k_s;
  if (wid >= 4) {
    float* dst = red + ntile * 256;
#pragma unroll
    for (int r = 0; r < 8; ++r) dst[(r + 8 * hlf) * 16 + ll] = acc[r];
  }
  __syncthreads();
  if (wid < 4) {
    const float* src = red + ntile * 256;
#pragma unroll
    for (int r = 0; r < 8; ++r) {
      const int M = r + 8 * hlf;
      const float v = acc[r] + src[M * 16 + ll];
      const int t = b * SEQ + i0 + M;                       // token index
      outH[(size_t)t * HIDDEN + h * HEAD_DIM + d0 + ll] = v;
    }
  }
}

// ---------------------------------------------------------------------------
// LayerNorm: one wave per token (512 elems -> 16 per lane), shuffle reductions
// ---------------------------------------------------------------------------
__global__ __launch_bounds__(256)
void layernorm_kernel(const float* __restrict__ res, const float* __restrict__ gamma,
                      const float* __restrict__ beta, float* __restrict__ out)
{
  const int wid  = threadIdx.x >> 5;
  const int lane = threadIdx.x & 31;
  const int t    = blockIdx.x * 8 + wid;
  const float* r = res + (size_t)t * HIDDEN;

  float vals[16];
  float s = 0.f, s2 = 0.f;
#pragma unroll
  for (int m = 0; m < 16; ++m) {
    const float v = r[lane + 32 * m];
    vals[m] = v; s += v; s2 += v * v;
  }
  s  = wave_sum(s);
  s2 = wave_sum(s2);
  const float mu  = s * (1.f / HIDDEN);
  const float var = s2 * (1.f / HIDDEN) - mu * mu;
  const float inv = rsqrtf(var + EPS_LN);

  float* o = out + (size_t)t * HIDDEN;
#pragma unroll
  for (int m = 0; m < 16; ++m) {
    const int j = lane + 32 * m;
    o[j] = (vals[m] - mu) * inv * gamma[j] + beta[j];
  }
}

// ---------------------------------------------------------------------------
extern "C" void kernel_launch(void* const* d_in, const int* in_sizes, int n_in,
                              void* d_out, int out_size, void* d_ws, size_t ws_size,
                              hipStream_t stream)
{
  const float* query = (const float*)d_in[0];
  const float* key_i = (const float*)d_in[1];
  const float* value = (const float*)d_in[2];
  const int*   mask  = (const int*)  d_in[3];
  const float* Wq    = (const float*)d_in[4];
  const float* bq    = (const float*)d_in[5];
  const float* Wk    = (const float*)d_in[6];
  const float* bk    = (const float*)d_in[7];
  const float* Wv    = (const float*)d_in[8];
  const float* bv    = (const float*)d_in[9];
  const float* Wo    = (const float*)d_in[10];
  const float* bo    = (const float*)d_in[11];
  const float* temp  = (const float*)d_in[12];
  const float* gamma = (const float*)d_in[13];
  const float* beta  = (const float*)d_in[14];

  float* out_main = (float*)d_out;                         // [NTOK, HIDDEN]
  float* attn_out = out_main + (size_t)NTOK * HIDDEN;      // [B,H,S,S]

  float* ws   = (float*)d_ws;
  const size_t SLAB = (size_t)NTOK * HIDDEN;               // 4.19M floats
  float* Qh   = ws;                                        // per-head [b][h][s][d]
  float* Kh   = Qh + SLAB;
  float* Vh   = Kh + SLAB;
  float* OutH = Vh + SLAB;                                 // merged-head [t][o]
  float* Res  = OutH + SLAB;                               // post-residual

  const dim3 gGemm(NTOK / 16 * (HIDDEN / 16) / 4);         // 4096 blocks, 4 waves each
  const dim3 bGemm(128);

  proj_gemm_kernel<<<gGemm, bGemm, 0, stream>>>(query, Wq, bq, nullptr, Qh, 0);
  proj_gemm_kernel<<<gGemm, bGemm, 0, stream>>>(key_i, Wk, bk, nullptr, Kh, 0);
  proj_gemm_kernel<<<gGemm, bGemm, 0, stream>>>(value, Wv, bv, nullptr, Vh, 0);

  const size_t smem = (size_t)(QT * QK_STRIDE + KT * QK_STRIDE + QT * SC_STRIDE)
                      * sizeof(float);                     // ~150 KB (<320 KB WGP LDS)
  l1_attention_kernel<<<dim3(BATCH * HEADS * (SEQ / QT)), dim3(256), smem, stream>>>(
      Qh, Kh, Vh, mask, temp, attn_out, OutH);

  proj_gemm_kernel<<<gGemm, bGemm, 0, stream>>>(OutH, Wo, bo, query, Res, 1);
  layernorm_kernel<<<dim3(NTOK / 8), dim3(256), 0, stream>>>(Res, gamma, beta, out_main);
}